// STPRI_2130303779455
// MI455X (gfx1250) — compile-verified
//
#include <hip/hip_runtime.h>
#include <hip/hip_bf16.h>

#define V 500
#define T 12
#define D 64
#define B 32

typedef __attribute__((ext_vector_type(2))) float v2f;
typedef __attribute__((ext_vector_type(4))) float v4f;
typedef __attribute__((ext_vector_type(8))) float v8f;

// ---------------------------------------------------------------------------
// Gaussian component row reduction: sum_d -0.5*((x[d]-mu[d])*is[d])^2
// ---------------------------------------------------------------------------
__device__ __forceinline__ float gc_row(const float* __restrict__ x,
                                        const float* __restrict__ mu,
                                        const float* __restrict__ is) {
    float acc = 0.f;
#pragma unroll
    for (int d = 0; d < D; ++d) {
        float t = (x[d] - mu[d]) * is[d];
        acc = __builtin_fmaf(-0.5f * t, t, acc);
    }
    return acc;
}

// ---------------------------------------------------------------------------
// Kernel A: tiny per-row reductions (V rows of sape twice; B*T combined scalars)
// ---------------------------------------------------------------------------
__global__ void prep_kernel(const float* __restrict__ sape,
                            const float* __restrict__ tape_i,
                            const float* __restrict__ tape_j,
                            const float* __restrict__ trpe,
                            const float* __restrict__ mus,
                            const float* __restrict__ invs,
                            float* __restrict__ sapei,
                            float* __restrict__ sapej,
                            float* __restrict__ svec) {
    int tid = threadIdx.x;
    if (tid < V) {
        sapei[tid] = gc_row(sape + tid * D, mus + 0 * D, invs + 0 * D);
        sapej[tid] = gc_row(sape + tid * D, mus + 1 * D, invs + 1 * D);
    }
    if (tid < B * T) {
        int b = tid / T;
        int t = tid % T;
        float ti = gc_row(tape_i + b * D,   mus + 2 * D, invs + 2 * D);
        float tj = gc_row(tape_j + tid * D, mus + 3 * D, invs + 3 * D);
        float tr = gc_row(trpe + t * D,     mus + 5 * D, invs + 5 * D);
        svec[tid] = ti + tj + tr;
    }
}

// ---------------------------------------------------------------------------
// Kernel B: srpe row-sums via V_WMMA_F32_16X16X4_F32.
// One wave32 per 16-row tile. A = elementwise Gaussian terms q (16x4 per op),
// B = all-ones, so D[m][n] accumulates the row sum of q in every column.
// Result written transposed with sapei/sapej folded in:
//   baseT[j*V + i] = srpe_g[i][j] + sapei[i] + sapej[j],  r = i*V + j.
// ---------------------------------------------------------------------------
__global__ void srpe_wmma_kernel(const float* __restrict__ srpe,
                                 const float* __restrict__ mus,
                                 const float* __restrict__ invs,
                                 const float* __restrict__ sapei,
                                 const float* __restrict__ sapej,
                                 float* __restrict__ baseT) {
    const int lane = threadIdx.x;          // 0..31 (wave32)
    const int r0   = blockIdx.x * 16;      // first global row of this tile
    const int half = lane >> 4;            // 0: K={0,1}, 1: K={2,3} slots
    const int mrow = lane & 15;            // A-matrix row M = lane % 16

    const float* __restrict__ mu = mus  + 4 * D;
    const float* __restrict__ is = invs + 4 * D;
    const float* __restrict__ rowp = srpe + (size_t)(r0 + mrow) * D;

    v8f c = {};
    v2f bones;
    bones.x = 1.0f;
    bones.y = 1.0f;

#pragma unroll
    for (int t = 0; t < 16; ++t) {
        const int d0 = 4 * t + half * 2;
        float x0 = __builtin_nontemporal_load(rowp + d0);
        float x1 = __builtin_nontemporal_load(rowp + d0 + 1);
        float u0 = (x0 - mu[d0])     * is[d0];
        float u1 = (x1 - mu[d0 + 1]) * is[d0 + 1];
        v2f a;
        a.x = -0.5f * u0 * u0;
        a.y = -0.5f * u1 * u1;
        // 8 args: (neg_a, A, neg_b, B, c_mod, C, reuse_a, reuse_b)
        c = __builtin_amdgcn_wmma_f32_16x16x4_f32(
            false, a, false, bones, (short)0, c, false, false);
    }

    // C layout: VGPR p, lanes 0-15 -> M=p ; lanes 16-31 -> M=8+p.
    // Lanes 0-7 write rows r0+0..7, lanes 16-23 write rows r0+8..15.
    const int p = lane & 7;
    if (((lane >> 3) & 1) == 0) {
        float rowsum = c[0];
#pragma unroll
        for (int q = 1; q < 8; ++q) rowsum = (p == q) ? c[q] : rowsum;
        int r = r0 + half * 8 + p;         // r = i*V + j
        int i = r / V;
        int j = r - i * V;
        baseT[(size_t)j * V + i] = rowsum + sapei[i] + sapej[j];
    }
}

// ---------------------------------------------------------------------------
// Kernel C: out[((b*V + j)*T + t)*V + i] = exp(baseT[j*V+i] + svec[b*T+t])
// float4 per thread; baseT (1 MB) stays hot in L2, output streamed NT.
// ---------------------------------------------------------------------------
__global__ void out_kernel(const v4f* __restrict__ baseT4,
                           const float* __restrict__ svec,
                           v4f* __restrict__ out4) {
    const int I4 = V / 4;                  // 125
    int idx = blockIdx.x * blockDim.x + threadIdx.x;
    int i4  = idx % I4;
    int rem = idx / I4;
    int t   = rem % T;
    rem    /= T;
    int j   = rem % V;
    int b   = rem / V;

    v4f bv  = baseT4[j * I4 + i4];
    float s = svec[b * T + t];

    v4f o;
    o.x = __expf(bv.x + s);
    o.y = __expf(bv.y + s);
    o.z = __expf(bv.z + s);
    o.w = __expf(bv.w + s);
    __builtin_nontemporal_store(o, out4 + idx);
}

// ---------------------------------------------------------------------------
extern "C" void kernel_launch(void* const* d_in, const int* in_sizes, int n_in,
                              void* d_out, int out_size, void* d_ws, size_t ws_size,
                              hipStream_t stream) {
    const float* sape   = (const float*)d_in[0];
    const float* tape_i = (const float*)d_in[1];
    const float* tape_j = (const float*)d_in[2];
    const float* srpe   = (const float*)d_in[3];
    const float* trpe   = (const float*)d_in[4];
    const float* mus    = (const float*)d_in[5];
    const float* invs   = (const float*)d_in[6];

    float* ws      = (float*)d_ws;
    float* baseT   = ws;                   // V*V floats (1 MB, 16B aligned)
    float* sapei_v = ws + V * V;           // V
    float* sapej_v = ws + V * V + V;       // V
    float* svec    = ws + V * V + 2 * V;   // B*T

    prep_kernel<<<1, 512, 0, stream>>>(sape, tape_i, tape_j, trpe, mus, invs,
                                       sapei_v, sapej_v, svec);

    srpe_wmma_kernel<<<(V * V) / 16, 32, 0, stream>>>(srpe, mus, invs,
                                                      sapei_v, sapej_v, baseT);

    const int total4 = B * V * T * (V / 4);   // 24,000,000 (exact multiple of 256)
    out_kernel<<<total4 / 256, 256, 0, stream>>>((const v4f*)baseT, svec,
                                                 (v4f*)d_out);
}